// NodeModel_74620761800880
// MI455X (gfx1250) — compile-verified
//
#include <hip/hip_runtime.h>
#include <hip/hip_bf16.h>

typedef __attribute__((ext_vector_type(2))) float v2f;
typedef __attribute__((ext_vector_type(8))) float v8f;

#define NEG_SLOPE 0.01f
#define NODE_EMB 128
#define EDGE_EMB 16
#define D_IN 160              // 128 + 2*16
#define MTILE 32              // rows per block (two 16-row WMMA tiles)
#define SA_STRIDE 164         // padded LDS row stride for A tile (avoid bank conflicts)
#define SH_STRIDE 132         // padded LDS row stride for h tile

// ---------------------------------------------------------------------------
// Kernel 1: zero the scatter accumulators in workspace
// ---------------------------------------------------------------------------
__global__ void zero_kernel(float* __restrict__ p, size_t n) {
    size_t i = (size_t)blockIdx.x * blockDim.x + threadIdx.x;
    if (i < n) p[i] = 0.0f;
}

// ---------------------------------------------------------------------------
// Kernel 2: edge scatter. 4 lanes per edge; each lane handles 4 features via
// one coalesced float4 load, then f32 atomics into L2-resident accumulators.
// ---------------------------------------------------------------------------
__global__ __launch_bounds__(256) void scatter_kernel(
    const int*   __restrict__ edge_index,  // [2, E]
    const float* __restrict__ edge_attr,   // [E, 16]
    float* __restrict__ sum_rec,           // [N, 16]
    float* __restrict__ sum_sent,          // [N, 16]
    float* __restrict__ cnt_rec,           // [N]
    float* __restrict__ cnt_sent,          // [N]
    int E)
{
    int t = blockIdx.x * 256 + threadIdx.x;
    int e = t >> 2;
    int q = t & 3;
    if (e >= E) return;

    int row = edge_index[e];              // receivers
    int col = edge_index[(size_t)E + e];  // senders

    const float4 v = *(const float4*)(edge_attr + (size_t)e * EDGE_EMB + q * 4);

    float* pr = sum_rec  + (size_t)row * EDGE_EMB + q * 4;
    float* ps = sum_sent + (size_t)col * EDGE_EMB + q * 4;
    atomicAdd(pr + 0, v.x); atomicAdd(pr + 1, v.y);
    atomicAdd(pr + 2, v.z); atomicAdd(pr + 3, v.w);
    atomicAdd(ps + 0, v.x); atomicAdd(ps + 1, v.y);
    atomicAdd(ps + 2, v.z); atomicAdd(ps + 3, v.w);
    if (q == 0) {
        atomicAdd(cnt_rec  + row, 1.0f);
        atomicAdd(cnt_sent + col, 1.0f);
    }
}

// ---------------------------------------------------------------------------
// Kernel 3: fused 2-layer MLP on 32-node tiles using V_WMMA_F32_16X16X4_F32.
// Block = 256 threads = 8 waves. Wave w owns output columns [16w, 16w+16)
// and runs TWO independent accumulator chains (rows 0-15 and 16-31) off the
// same B operand, halving W traffic and hiding WMMA accumulate latency.
// ---------------------------------------------------------------------------
__global__ __launch_bounds__(256) void mlp_wmma_kernel(
    const float* __restrict__ x,         // [N, 128]
    const float* __restrict__ sum_rec,   // [N, 16]
    const float* __restrict__ sum_sent,  // [N, 16]
    const float* __restrict__ cnt_rec,   // [N]
    const float* __restrict__ cnt_sent,  // [N]
    const float* __restrict__ W1,        // [160, 128] row-major
    const float* __restrict__ b1,        // [128]
    const float* __restrict__ W2,        // [128, 128] row-major
    const float* __restrict__ b2,        // [128]
    float* __restrict__ out,             // [N, 128]
    int N)
{
    __shared__ float sA[MTILE * SA_STRIDE];   // concat(x, rec_mean, sent_mean) tile
    __shared__ float sH[MTILE * SH_STRIDE];   // hidden-layer tile

    const int tid = threadIdx.x;
    const int n0  = blockIdx.x * MTILE;

    // ---- stage A tile: x part (32x128) ----
    for (int i = tid; i < MTILE * NODE_EMB; i += 256) {
        int r = i >> 7, c = i & 127;
        int rg = n0 + r; if (rg >= N) rg = N - 1;   // clamp (32 divides N=100000)
        sA[r * SA_STRIDE + c] = x[(size_t)rg * NODE_EMB + c];
    }
    // ---- stage A tile: scatter means (32x16 each), fuse the division ----
    for (int i = tid; i < MTILE * EDGE_EMB; i += 256) {
        int r = i >> 4, c = i & 15;
        int rg = n0 + r; if (rg >= N) rg = N - 1;
        float cr = cnt_rec[rg];  cr = cr > 1.0f ? cr : 1.0f;
        float cs = cnt_sent[rg]; cs = cs > 1.0f ? cs : 1.0f;
        sA[r * SA_STRIDE + NODE_EMB + c]            = sum_rec[(size_t)rg * EDGE_EMB + c] / cr;
        sA[r * SA_STRIDE + NODE_EMB + EDGE_EMB + c] = sum_sent[(size_t)rg * EDGE_EMB + c] / cs;
    }
    __syncthreads();

    const int lane    = tid & 31;
    const int wv      = tid >> 5;            // 0..7 -> 16-wide column slice
    const int m       = lane & 15;           // A-row / B-col within tile
    const int khalf   = (lane >> 4) << 1;    // lanes 0-15: K 0,1 ; lanes 16-31: K 2,3
    const int colBase = wv * 16;
    const int rowOff  = (lane >> 4) << 3;    // C/D: M = vgpr + 8*(lane/16)

    // ================= GEMM1: [32,160] @ [160,128] =================
    v8f acc0 = {};   // rows  0..15 of the block tile
    v8f acc1 = {};   // rows 16..31 of the block tile
    #pragma unroll 8
    for (int k = 0; k < D_IN; k += 4) {
        v2f a0, a1, b;
        b.x  = W1[(size_t)(k + khalf)     * NODE_EMB + colBase + m];
        b.y  = W1[(size_t)(k + khalf + 1) * NODE_EMB + colBase + m];
        a0.x = sA[m * SA_STRIDE + k + khalf];
        a0.y = sA[m * SA_STRIDE + k + khalf + 1];
        a1.x = sA[(16 + m) * SA_STRIDE + k + khalf];
        a1.y = sA[(16 + m) * SA_STRIDE + k + khalf + 1];
        acc0 = __builtin_amdgcn_wmma_f32_16x16x4_f32(
                   false, a0, false, b, (short)0, acc0, false, false);
        acc1 = __builtin_amdgcn_wmma_f32_16x16x4_f32(
                   false, a1, false, b, (short)0, acc1, false, false);
    }

    // bias + LeakyReLU on accumulator registers, spill h tile to LDS
    {
        const float bias = b1[colBase + m];
        #pragma unroll
        for (int i = 0; i < 8; ++i) {
            float v0 = acc0[i] + bias;
            v0 = v0 >= 0.0f ? v0 : NEG_SLOPE * v0;
            sH[(i + rowOff) * SH_STRIDE + colBase + m] = v0;
            float v1 = acc1[i] + bias;
            v1 = v1 >= 0.0f ? v1 : NEG_SLOPE * v1;
            sH[(16 + i + rowOff) * SH_STRIDE + colBase + m] = v1;
        }
    }
    __syncthreads();

    // ================= GEMM2: [32,128] @ [128,128] =================
    v8f acc2 = {};
    v8f acc3 = {};
    #pragma unroll 8
    for (int k = 0; k < NODE_EMB; k += 4) {
        v2f a0, a1, b;
        b.x  = W2[(size_t)(k + khalf)     * NODE_EMB + colBase + m];
        b.y  = W2[(size_t)(k + khalf + 1) * NODE_EMB + colBase + m];
        a0.x = sH[m * SH_STRIDE + k + khalf];
        a0.y = sH[m * SH_STRIDE + k + khalf + 1];
        a1.x = sH[(16 + m) * SH_STRIDE + k + khalf];
        a1.y = sH[(16 + m) * SH_STRIDE + k + khalf + 1];
        acc2 = __builtin_amdgcn_wmma_f32_16x16x4_f32(
                   false, a0, false, b, (short)0, acc2, false, false);
        acc3 = __builtin_amdgcn_wmma_f32_16x16x4_f32(
                   false, a1, false, b, (short)0, acc3, false, false);
    }

    const float bias2 = b2[colBase + m];
    #pragma unroll
    for (int i = 0; i < 8; ++i) {
        int rg0 = n0 + i + rowOff;
        if (rg0 < N)
            out[(size_t)rg0 * NODE_EMB + colBase + m] = acc2[i] + bias2;
        int rg1 = n0 + 16 + i + rowOff;
        if (rg1 < N)
            out[(size_t)rg1 * NODE_EMB + colBase + m] = acc3[i] + bias2;
    }
}

// ---------------------------------------------------------------------------
// Host launcher
// ---------------------------------------------------------------------------
extern "C" void kernel_launch(void* const* d_in, const int* in_sizes, int n_in,
                              void* d_out, int out_size, void* d_ws, size_t ws_size,
                              hipStream_t stream) {
    const float* x  = (const float*)d_in[0];
    const int*   ei = (const int*)  d_in[1];
    const float* ea = (const float*)d_in[2];
    const float* W1 = (const float*)d_in[3];
    const float* b1 = (const float*)d_in[4];
    const float* W2 = (const float*)d_in[5];
    const float* b2 = (const float*)d_in[6];
    float* out = (float*)d_out;

    const int N = in_sizes[0] / NODE_EMB;
    const int E = in_sizes[1] / 2;

    float* ws       = (float*)d_ws;
    float* sum_rec  = ws;                       // N*16
    float* sum_sent = ws + (size_t)N * 16;      // N*16
    float* cnt_rec  = ws + (size_t)N * 32;      // N
    float* cnt_sent = ws + (size_t)N * 33;      // N

    // 1) zero accumulators
    size_t zn = (size_t)N * 34;
    zero_kernel<<<(int)((zn + 255) / 256), 256, 0, stream>>>(ws, zn);

    // 2) edge scatter (4 lanes per edge)
    long long st = 4LL * E;
    scatter_kernel<<<(int)((st + 255) / 256), 256, 0, stream>>>(
        ei, ea, sum_rec, sum_sent, cnt_rec, cnt_sent, E);

    // 3) fused MLP with f32 WMMA, 32-row tiles, dual accumulator chains
    mlp_wmma_kernel<<<(N + MTILE - 1) / MTILE, 256, 0, stream>>>(
        x, sum_rec, sum_sent, cnt_rec, cnt_sent, W1, b1, W2, b2, out, N);
}